// Model_16252156248550
// MI455X (gfx1250) — compile-verified
//
#include <hip/hip_runtime.h>

typedef __attribute__((ext_vector_type(2))) float v2f;
typedef __attribute__((ext_vector_type(8))) float v8f;
typedef int   v4i __attribute__((vector_size(16)));   // builtin pointee type
typedef float v4f __attribute__((vector_size(16)));   // native vector for NT ld/st

#define B_    32
#define T_    64
#define C_    2
#define H_    32
#define W_    32
#define NK_   16
#define KS_   6
#define ACT_  27
#define POS_  729          // ACT*ACT
#define FEAT_ 11664        // POS*NK
#define RU_   16
#define KCONV 72           // C*KS*KS
#define PADW  73           // weight LDS row stride; 73%64==9 -> conflict-free

#define CHUNK 128
#define PAD   132          // LDS row stride (floats); 132%64==4 -> conflict-free A gathers
#define NCH   92           // ceil(11664/128); last chunk has 16 feats

// gfx1250 async global->LDS path (ASYNCcnt); fall back to sync copy if the
// toolchain doesn't declare the builtins (also keeps the host pass happy).
#if __has_builtin(__builtin_amdgcn_global_load_async_to_lds_b128) && \
    __has_builtin(__builtin_amdgcn_s_wait_asynccnt)
#define HAVE_ASYNC_LDS 1
#endif

// ---------------------------------------------------------------------------
// zero the atomic accumulation buffer (must run every launch: atomics add)
__global__ void zero_pre_kernel(float* __restrict__ p, int n) {
    int i = blockIdx.x * blockDim.x + threadIdx.x;
    if (i < n) p[i] = 0.0f;
}

// ---------------------------------------------------------------------------
// Conv (valid, 6x6, C=2 -> NK=16) as implicit GEMM per frame, TRANSPOSED:
//   A = weights  (M = 16 channels, K = 72)
//   B = patches  (K = 72, N = 16 spatial positions)  -> 46 position tiles
// D lanes = consecutive positions => coalesced stores. f32 WMMA 16x16x4.
// out_conv is streamed with non-temporal stores (never re-read) so out_flat
// can stay resident in the 192MB L2 for the following kernel.
__global__ __launch_bounds__(256) void conv_wmma_kernel(
    const float* __restrict__ x, const float* __restrict__ conv_w,
    const float* __restrict__ conv_b,
    float* __restrict__ out_conv, float* __restrict__ out_flat)
{
    __shared__ float sIn[C_ * H_ * W_];   // 2048 floats (frame)
    __shared__ float sWa[NK_ * PADW];     // sWa[ch*73 + k] = conv_w[ch*72 + k]
    __shared__ int   sOff[KCONV];         // patch gather offset per k

    const int frame = blockIdx.x;         // b*T + t
    const int tid   = threadIdx.x;

    // stage input frame into LDS (async DMA on gfx1250; tracked by ASYNCcnt)
    const v4f* src4 = (const v4f*)(x + (size_t)frame * (C_ * H_ * W_));
    v4f* dst4 = (v4f*)sIn;
#ifdef HAVE_ASYNC_LDS
    for (int i = tid; i < (C_ * H_ * W_) / 4; i += 256)
        __builtin_amdgcn_global_load_async_to_lds_b128(
            (__attribute__((address_space(1))) v4i*)(src4 + i),
            (__attribute__((address_space(3))) v4i*)(dst4 + i), 0, 0);
#else
    for (int i = tid; i < (C_ * H_ * W_) / 4; i += 256)
        dst4[i] = __builtin_nontemporal_load(src4 + i);
#endif
    // stage weights as A fragments (natural [ch][k] layout, padded rows)
    for (int i = tid; i < NK_ * KCONV; i += 256) {
        int ch = i / KCONV, k = i - ch * KCONV;
        sWa[ch * PADW + k] = conv_w[i];
    }
    if (tid < KCONV) {
        int c = tid / 36, r = (tid % 36) / 6, q = tid % 6;
        sOff[tid] = c * (H_ * W_) + r * W_ + q;
    }
#ifdef HAVE_ASYNC_LDS
    __builtin_amdgcn_s_wait_asynccnt(0);
#endif
    __syncthreads();

    const int lane = tid & 31;
    const int wave = tid >> 5;
    const int hf   = lane >> 4;           // half-wave select
    const int l16  = lane & 15;
    const int kOff = hf * 2;

    float bias8[8];                        // per D-row (= channel) bias
    #pragma unroll
    for (int r = 0; r < 8; ++r) bias8[r] = conv_b[hf * 8 + r];

    const size_t frameOut = (size_t)frame * FEAT_;
    for (int tile = wave; tile < 46; tile += 8) {
        const int n0  = tile * 16;
        const int pos = n0 + l16;                       // this lane's position
        const int pc  = pos > POS_ - 1 ? POS_ - 1 : pos;
        const int base = (pc / ACT_) * W_ + (pc % ACT_);
        v8f acc = {};
        #pragma unroll
        for (int s = 0; s < 18; ++s) {
            const int k = 4 * s + kOff;
            v2f a, b;
            a.x = sWa[l16 * PADW + k];      // A: row = channel, K per half
            a.y = sWa[l16 * PADW + k + 1];
            b.x = sIn[base + sOff[k]];      // B: col = position, K per half
            b.y = sIn[base + sOff[k + 1]];
            acc = __builtin_amdgcn_wmma_f32_16x16x4_f32(
                false, a, false, b, (short)0, acc, false, false);
        }
        // D: lane = position column; VGPR r = channel hf*8 + r
        if (pos < POS_) {
            #pragma unroll
            for (int r = 0; r < 8; ++r) {
                const size_t idx = frameOut + (size_t)(hf * 8 + r) * POS_ + pos;
                float v = acc[r] + bias8[r];
                __builtin_nontemporal_store(v, &out_conv[idx]);  // never re-read
                out_flat[idx] = fmaxf(v, 0.0f);                  // keep in L2
            }
        }
    }
}

// ---------------------------------------------------------------------------
// Fused AdaptiveLayer scan + w_ih projection.
// Workgroup = (batch b, feature-chunk of 128). Chain runs over T in LDS,
// then WMMA GEMM:  pre[t, j] += sum_f adapted[t, f] * w_ih[j, f]
// M = 64 (time, 4 tiles x 4 waves), N = 16, K = cnt. fp32 atomics accumulate.
__global__ __launch_bounds__(128) void adapt_gemm_kernel(
    const float* __restrict__ out_flat, const float* __restrict__ w_ih,
    float* __restrict__ preBuf)
{
    __shared__ float sA[T_ * PAD];        // adapted tile, 33 KB
    __shared__ float sW[CHUNK * 16];      // w_ih fragment, 8 KB

    const int b   = blockIdx.y;
    const int f0  = blockIdx.x * CHUNK;
    int cnt = FEAT_ - f0; if (cnt > CHUNK) cnt = CHUNK;   // 128 or 16 (both %4==0)
    const int tid = threadIdx.x;

    // sW[kk*16 + j] = w_ih[j*FEAT + f0 + kk]
    for (int i = tid; i < cnt * 16; i += 128) {
        int j = i / cnt, kk = i - j * cnt;
        sW[kk * 16 + j] = w_ih[(size_t)j * FEAT_ + f0 + kk];
    }
    // stage relu'd conv features (expected L2-resident) for all 64 frames
    for (int i = tid; i < T_ * cnt; i += 128) {
        int t = i / cnt, kk = i - t * cnt;
        sA[t * PAD + kk] = out_flat[((size_t)(b * T_ + t)) * FEAT_ + f0 + kk];
    }
    __syncthreads();

    // per-feature adaptation chain over time, in place in LDS
    if (tid < cnt) {
        float ad = 0.0f;
        #pragma unroll 4
        for (int t = 0; t < T_; ++t) {
            float v = sA[t * PAD + tid];
            float o = fmaxf(v - ad, 0.0f);
            ad = (ad + 0.1f * o) * 0.9f;
            sA[t * PAD + tid] = o;
        }
    }
    __syncthreads();

    const int lane = tid & 31;
    const int wave = tid >> 5;            // 4 waves = 4 time-tiles
    const int hf   = lane >> 4;
    const int l16  = lane & 15;
    const int kOff = hf * 2;
    const int t0   = wave * 16;
    const int trow = t0 + l16;
    const int steps = cnt >> 2;

    v8f acc = {};
    for (int s = 0; s < steps; ++s) {
        const int k = 4 * s + kOff;
        v2f a, b;
        a.x = sA[trow * PAD + k];
        a.y = sA[trow * PAD + k + 1];
        b.x = sW[k * 16 + l16];
        b.y = sW[(k + 1) * 16 + l16];
        acc = __builtin_amdgcn_wmma_f32_16x16x4_f32(
            false, a, false, b, (short)0, acc, false, false);
    }
    // accumulate into preBuf[(b*T + t)*16 + j] (lanes -> consecutive j)
    const int tBase = t0 + hf * 8;
    #pragma unroll
    for (int r = 0; r < 8; ++r)
        atomicAdd(&preBuf[((size_t)(b * T_ + tBase + r)) * 16 + l16], acc[r]);
}

// ---------------------------------------------------------------------------
// Sequential SimpleRNN (+adaptation) and FC head. One workgroup, 512 threads:
// thread = (batch b, unit j). h lives in LDS, adapt in registers.
__global__ __launch_bounds__(512) void rnn_fc_kernel(
    const float* __restrict__ preBuf, const float* __restrict__ w_hh,
    const float* __restrict__ b_ih, const float* __restrict__ b_hh,
    const float* __restrict__ fc_w, const float* __restrict__ fc_b,
    float* __restrict__ out_rnn, float* __restrict__ out_fc)
{
    __shared__ float hLds[B_ * RU_];
    const int tid = threadIdx.x;
    const int b = tid >> 4, j = tid & 15;

    float whh[16];
    #pragma unroll
    for (int k = 0; k < 16; ++k) whh[k] = w_hh[j * 16 + k];
    const float bias = b_ih[j] + b_hh[j];

    float fcw[16]; float fcb = 0.0f;
    if (j < 2) {
        fcb = fc_b[j];
        #pragma unroll
        for (int k = 0; k < 16; ++k) fcw[k] = fc_w[j * 16 + k];
    }

    float adapt = 0.0f;
    hLds[tid] = 0.0f;
    __syncthreads();

    for (int t = 0; t < T_; ++t) {
        float s = preBuf[((size_t)(b * T_ + t)) * 16 + j] + bias;
        #pragma unroll
        for (int k = 0; k < 16; ++k) s += hLds[b * 16 + k] * whh[k];
        float a = fmaxf(s - adapt, 0.0f);       // h = relu(a) = a
        adapt = (adapt + 0.4f * a) * 0.9f;
        out_rnn[((size_t)(b * T_ + t)) * 16 + j] = a;
        __syncthreads();
        hLds[tid] = a;
        __syncthreads();
        if (j < 2) {
            float acc = fcb;
            #pragma unroll
            for (int k = 0; k < 16; ++k) acc += hLds[b * 16 + k] * fcw[k];
            out_fc[((size_t)(b * T_ + t)) * 2 + j] = acc;
        }
    }
}

// ---------------------------------------------------------------------------
extern "C" void kernel_launch(void* const* d_in, const int* in_sizes, int n_in,
                              void* d_out, int out_size, void* d_ws, size_t ws_size,
                              hipStream_t stream) {
    const float* x      = (const float*)d_in[0];
    const float* conv_w = (const float*)d_in[1];
    const float* conv_b = (const float*)d_in[2];
    const float* w_ih   = (const float*)d_in[3];
    const float* w_hh   = (const float*)d_in[4];
    const float* b_ih   = (const float*)d_in[5];
    const float* b_hh   = (const float*)d_in[6];
    const float* fc_w   = (const float*)d_in[7];
    const float* fc_b   = (const float*)d_in[8];

    float* out_fc   = (float*)d_out;                              // [B,T,2]
    float* out_conv = out_fc + (size_t)B_ * T_ * 2;               // [B,T,16,27,27]
    float* out_flat = out_conv + (size_t)B_ * T_ * FEAT_;         // [B,T,FEAT]
    float* out_rnn  = out_flat + (size_t)B_ * T_ * FEAT_;         // [B,T,16]

    float* preBuf = (float*)d_ws;                                 // [B,T,16] fp32

    const int nPre = B_ * T_ * RU_;
    zero_pre_kernel<<<(nPre + 255) / 256, 256, 0, stream>>>(preBuf, nPre);

    conv_wmma_kernel<<<B_ * T_, 256, 0, stream>>>(x, conv_w, conv_b,
                                                  out_conv, out_flat);

    dim3 g2(NCH, B_);
    adapt_gemm_kernel<<<g2, 128, 0, stream>>>(out_flat, w_ih, preBuf);

    rnn_fc_kernel<<<1, 512, 0, stream>>>(preBuf, w_hh, b_ih, b_hh,
                                         fc_w, fc_b, out_rnn, out_fc);
}